// ResampleLayer_22007412424982
// MI455X (gfx1250) — compile-verified
//
#include <hip/hip_runtime.h>

// Problem constants from the reference
#define B_N      8192
#define REST_LEN 4500
#define OUT_DIM  2048
#define MIN_W    3550.0f
#define MAX_W    10400.0f

typedef __attribute__((ext_vector_type(4))) unsigned int uint32x4;
typedef __attribute__((ext_vector_type(8))) int          int32x8;
typedef __attribute__((ext_vector_type(4))) int          int32x4;

// rest = linspace(MIN_W/2.2, MAX_W/1.5, REST_LEN) is affine:
// rest[i] = A + i*S.  argmin_i |rest[i] - t| == clamped nearest index.
// Resolve the floor/ceil candidate pair with the same |.| compare the
// reference performs (tie -> lower index, matching argmin's first-occurrence).
__device__ __forceinline__ int nearest_idx(float t) {
    const float A = MIN_W / 2.2f;
    const float S = (MAX_W / 1.5f - MIN_W / 2.2f) / (float)(REST_LEN - 1);
    float u  = (t - A) / S;
    int   i0 = (int)floorf(u);
    i0 = max(0, min(i0, REST_LEN - 1));
    int   i1 = min(i0 + 1, REST_LEN - 1);
    float d0 = fabsf(fmaf((float)i0, S, A) - t);
    float d1 = fabsf(fmaf((float)i1, S, A) - t);
    return (d1 < d0) ? i1 : i0;
}

__global__ __launch_bounds__(256) void
ResampleLayer_22007412424982_kernel(const float* __restrict__ x,
                                    const float* __restrict__ z,
                                    float* __restrict__ out) {
    // Window is at most ~4.4k floats; round LDS buffer up generously.
    __shared__ float tile[4608];

    const int b = blockIdx.x;

    // ---- per-row bounds (uniform across the block) ----
    const float div_ = 1.0f + z[b];
    int start = nearest_idx(MIN_W / div_);
    int stop  = nearest_idx(MAX_W / div_);
    // start >= 0 and stop <= 4499 hold by construction (clamped above)
    start = __builtin_amdgcn_readfirstlane(start);   // force SGPR / uniform
    stop  = __builtin_amdgcn_readfirstlane(stop);
    int length = max(stop - start, 1);
    length = min(length, REST_LEN - start);          // defensive in-row clamp

    const float* src = x + (size_t)b * REST_LEN + start;

    // ---- TDM: DMA the contiguous window HBM -> LDS (one op per block) ----
    if (threadIdx.x < 32u) {                         // wave 0 only (wave32)
        const unsigned n        = (unsigned)length;            // elements (fp32)
        const unsigned long long gaddr = (unsigned long long)(uintptr_t)src;
        const unsigned lds_off  = (unsigned)(size_t)(void*)tile; // low 32b = LDS byte offset

        // D# group 0: [1:0]=count=1 | lds_addr | global_addr[56:0] | type=2
        uint32x4 g0;
        g0.x = 1u;
        g0.y = lds_off;
        g0.z = (unsigned)(gaddr & 0xffffffffull);
        g0.w = ((unsigned)(gaddr >> 32) & 0x01ffffffu) | (2u << 30);

        // D# group 1:
        //  d0: workgroup_mask=0, data_size=2 (4 bytes), no barrier/iterate/pad
        //  d1[31:16] = tensor_dim0[15:0]
        //  d2[15:0]  = tensor_dim0[31:16],  d2[31:16] = tensor_dim1[15:0] = 1
        //  d3[31:16] = tile_dim0 = n
        //  d4[15:0]  = tile_dim1 = 1, tile_dim2 = 0
        //  d5        = tensor_dim0_stride low 32 = REST_LEN (single row: unused)
        int32x8 g1;
        g1[0] = 0x00020000;
        g1[1] = (int)((n & 0xffffu) << 16);
        g1[2] = (int)(((n >> 16) & 0xffffu) | (1u << 16));
        g1[3] = (int)((n & 0xffffu) << 16);
        g1[4] = 1;
        g1[5] = REST_LEN;
        g1[6] = 0;
        g1[7] = 0;

        int32x4 g2 = {0, 0, 0, 0};            // dims 2..4 unused (normal mode)
        int32x4 g3 = {0, 0, 0, 0};
        int32x8 g4 = {0, 0, 0, 0, 0, 0, 0, 0}; // extra operand on 6-arg toolchain

        __builtin_amdgcn_tensor_load_to_lds(g0, g1, g2, g3, g4, /*cpol=*/0);
        __builtin_amdgcn_s_wait_tensorcnt(0);
    }
    __syncthreads();                                  // publish LDS to all 8 waves

    // ---- resample: vectorized coalesced stores, LDS gathers ----
    const float lenf  = (float)length;
    const float scale = lenf * (1.0f / (float)OUT_DIM);   // length / OUT_DIM
    const float lenm1 = lenf - 1.0f;
    float* __restrict__ outRow = out + (size_t)b * OUT_DIM;

    // Each thread produces 4 consecutive outputs per iteration -> b128 stores;
    // a wave covers 512 contiguous bytes per store issue.
#pragma unroll
    for (int k = 0; k < OUT_DIM / (256 * 4); ++k) {
        const int j0 = ((int)threadIdx.x + k * 256) * 4;
        float4 r;
#pragma unroll
        for (int e = 0; e < 4; ++e) {
            const float pos  = ((float)(j0 + e) + 0.5f) * scale - 0.5f;
            const float f    = floorf(pos);
            const float lerp = pos - f;
            const int   low  = (int)fminf(fmaxf(f, 0.0f), lenm1);
            const int   high = (int)fminf(fmaxf(ceilf(pos), 0.0f), lenm1);
            const float gl   = tile[low];
            const float gh   = tile[high];
            (&r.x)[e] = gl * (1.0f - lerp) + gh * lerp;
        }
        *(float4*)(outRow + j0) = r;
    }
}

extern "C" void kernel_launch(void* const* d_in, const int* in_sizes, int n_in,
                              void* d_out, int out_size, void* d_ws, size_t ws_size,
                              hipStream_t stream) {
    (void)in_sizes; (void)n_in; (void)d_ws; (void)ws_size; (void)out_size;
    const float* x = (const float*)d_in[0];   // (B, 4500, 1) fp32
    const float* z = (const float*)d_in[1];   // (B, 1)       fp32
    float* out = (float*)d_out;               // (B, 2048, 1) fp32

    ResampleLayer_22007412424982_kernel<<<B_N, 256, 0, stream>>>(x, z, out);
}